// GAT_48241072669151
// MI455X (gfx1250) — compile-verified
//
#include <hip/hip_runtime.h>
#include <hip/hip_bf16.h>

typedef __attribute__((ext_vector_type(2))) float v2f;
typedef __attribute__((ext_vector_type(8))) float v8f;

#define NEG_SLOPE 0.2f

// ---------------------------------------------------------------------------
// C[M,Nc] = A[M,K] @ B[K,Nc], fp32, via V_WMMA_F32_16X16X4_F32.
// One wave computes a 16x16 C tile; 8 waves/block share a K x 16 B tile in LDS.
// K must be 256 (both layers). M must be a multiple of 16 (50000 = 3125*16).
// A-frag layout (ISA 7.12.2, 32-bit A 16x4): lanes 0-15 hold K={0,1},
// lanes 16-31 hold K={2,3}; M = lane&15.  B/C striped across lanes per VGPR.
// ---------------------------------------------------------------------------
__global__ void gat_gemm_wmma_f32(const float* __restrict__ A,
                                  const float* __restrict__ B,
                                  float* __restrict__ C,
                                  int M, int K, int Nc) {
    __shared__ float Bs[256 * 16];           // 16 KB: B[:, ncol0..ncol0+16)
    const int ncol0 = blockIdx.x * 16;

    for (int idx = threadIdx.x; idx < K * 16; idx += blockDim.x) {
        int kk = idx >> 4;
        int nn = idx & 15;
        Bs[idx] = B[(size_t)kk * Nc + ncol0 + nn];
    }
    __syncthreads();

    const int wave  = threadIdx.x >> 5;
    const int lane  = threadIdx.x & 31;
    const int mtile = blockIdx.y * 8 + wave;
    if (mtile * 16 >= M) return;

    const int mbase = mtile * 16;
    const int hi    = lane >> 4;             // 0: K pair {0,1}; 1: K pair {2,3}
    const int nl    = lane & 15;
    const int row   = mbase + nl;            // A-matrix M index for this lane
    const int koff  = hi * 2;

    v8f c = {};
    for (int k = 0; k < K; k += 4) {
        const float2 av = *(const float2*)(A + (size_t)row * K + k + koff);
        v2f a; a.x = av.x; a.y = av.y;
        v2f b;
        b.x = Bs[(k + koff) * 16 + nl];
        b.y = Bs[(k + koff + 1) * 16 + nl];
        c = __builtin_amdgcn_wmma_f32_16x16x4_f32(
                /*neg_a=*/false, a, /*neg_b=*/false, b,
                /*c_mod=*/(short)0, c, /*reuse_a=*/false, /*reuse_b=*/false);
    }

#pragma unroll
    for (int g = 0; g < 8; ++g) {
        // C/D layout: VGPR g -> M = g (lanes 0-15) or M = g+8 (lanes 16-31)
        C[(size_t)(mbase + g + hi * 8) * Nc + ncol0 + nl] = c[g];
    }
}

// ---------------------------------------------------------------------------
// a_src[n,h] = <h[n,h,:], att_src[h,:]> ; a_dst likewise.  Thread per (n,h).
// ---------------------------------------------------------------------------
__global__ void gat_attn_coef(const float* __restrict__ h,
                              const float* __restrict__ att_src,
                              const float* __restrict__ att_dst,
                              float* __restrict__ a_src,
                              float* __restrict__ a_dst,
                              int Nn, int H, int Fh) {
    int idx = blockIdx.x * blockDim.x + threadIdx.x;
    if (idx >= Nn * H) return;
    int n  = idx / H;
    int hd = idx - n * H;
    const float* hp  = h + (size_t)n * H * Fh + hd * Fh;
    const float* asp = att_src + hd * Fh;
    const float* adp = att_dst + hd * Fh;
    float s = 0.f, d = 0.f;
    for (int f = 0; f < Fh; ++f) {
        float v = hp[f];
        s += v * asp[f];
        d += v * adp[f];
    }
    a_src[idx] = s;
    a_dst[idx] = d;
}

// ---------------------------------------------------------------------------
// Softmax denominator: denom[dst,h] += exp(leaky(a_src[src,h]+a_dst[dst,h])).
// Logits are O(1) here, so no max-subtraction is needed numerically.
// Edge e in [0,E): src=ei[e], dst=ei[E+e]; e in [E,EP): self-loop e-E.
// ---------------------------------------------------------------------------
__global__ void gat_edge_denom(const int* __restrict__ ei, int E, int EP,
                               const float* __restrict__ a_src,
                               const float* __restrict__ a_dst,
                               float* __restrict__ denom, int H) {
    int idx = blockIdx.x * blockDim.x + threadIdx.x;
    if (idx >= EP * H) return;
    int e  = idx / H;
    int hd = idx - e * H;
    int s, d;
    if (e < E) { s = ei[e]; d = ei[E + e]; } else { s = e - E; d = e - E; }
    float v = a_src[s * H + hd] + a_dst[d * H + hd];
    v = v > 0.f ? v : NEG_SLOPE * v;
    atomicAdd(&denom[d * H + hd], expf(v));
}

// ---------------------------------------------------------------------------
// Aggregation: one wave per edge; 32 lanes sweep the H*Fh features in
// contiguous chunks (layer1: chunk == one head).  alpha is recomputed
// (exp is cheap) instead of spilling a per-(edge,head) buffer.
// ---------------------------------------------------------------------------
__global__ void gat_edge_aggregate(const int* __restrict__ ei, int E, int EP,
                                   const float* __restrict__ a_src,
                                   const float* __restrict__ a_dst,
                                   const float* __restrict__ denom,
                                   const float* __restrict__ h,
                                   float* __restrict__ out,
                                   int H, int Fh) {
    int gwave = (blockIdx.x * blockDim.x + threadIdx.x) >> 5;
    int lane  = threadIdx.x & 31;
    if (gwave >= EP) return;
    int e = gwave;
    int s, d;
    if (e < E) { s = ei[e]; d = ei[E + e]; } else { s = e - E; d = e - E; }
    const int HF = H * Fh;
    for (int c = lane; c < HF; c += 32) {
        int hd = c / Fh;
        float v = a_src[s * H + hd] + a_dst[d * H + hd];
        v = v > 0.f ? v : NEG_SLOPE * v;
        float alpha = expf(v) / denom[d * H + hd];
        atomicAdd(&out[(size_t)d * HF + c], alpha * h[(size_t)s * HF + c]);
    }
}

__global__ void gat_bias_relu(float* __restrict__ y, const float* __restrict__ b,
                              int total, int C) {
    int i = blockIdx.x * blockDim.x + threadIdx.x;
    if (i >= total) return;
    float v = y[i] + b[i % C];
    y[i] = v > 0.f ? v : 0.f;
}

__global__ void gat_bias_add(float* __restrict__ y, const float* __restrict__ b,
                             int total, int C) {
    int i = blockIdx.x * blockDim.x + threadIdx.x;
    if (i >= total) return;
    y[i] = y[i] + b[i % C];
}

// ---------------------------------------------------------------------------
extern "C" void kernel_launch(void* const* d_in, const int* in_sizes, int n_in,
                              void* d_out, int out_size, void* d_ws, size_t ws_size,
                              hipStream_t stream) {
    (void)n_in; (void)out_size; (void)ws_size;

    const float* x    = (const float*)d_in[0];
    const int*   ei   = (const int*)d_in[1];
    const float* W1   = (const float*)d_in[2];
    const float* as1w = (const float*)d_in[3];
    const float* ad1w = (const float*)d_in[4];
    const float* b1   = (const float*)d_in[5];
    const float* W2   = (const float*)d_in[6];
    const float* as2w = (const float*)d_in[7];
    const float* ad2w = (const float*)d_in[8];
    const float* b2   = (const float*)d_in[9];

    const int IN  = 256, HID = 32, HEADS = 8, OUT = 64;
    const int Nn  = in_sizes[0] / IN;      // 50000
    const int E   = in_sizes[1] / 2;       // 800000
    const int EP  = E + Nn;                // with self-loops

    // Workspace carve-out (all fp32).
    float* p = (float*)d_ws;
    float* h1     = p; p += (size_t)Nn * (HEADS * HID);
    float* a_src1 = p; p += (size_t)Nn * HEADS;
    float* a_dst1 = p; p += (size_t)Nn * HEADS;
    float* den1   = p; p += (size_t)Nn * HEADS;
    float* agg1   = p; p += (size_t)Nn * (HEADS * HID);
    float* h2     = p; p += (size_t)Nn * OUT;
    float* a_src2 = p; p += (size_t)Nn;
    float* a_dst2 = p; p += (size_t)Nn;
    float* den2   = p; p += (size_t)Nn;
    float* outF   = (float*)d_out;

    // Zero accumulators (graph-capturable memset nodes).
    hipMemsetAsync(den1, 0, (size_t)Nn * HEADS * sizeof(float), stream);
    hipMemsetAsync(agg1, 0, (size_t)Nn * HEADS * HID * sizeof(float), stream);
    hipMemsetAsync(den2, 0, (size_t)Nn * sizeof(float), stream);
    hipMemsetAsync(outF, 0, (size_t)Nn * OUT * sizeof(float), stream);

    const int Mtiles = (Nn + 15) / 16;     // 3125
    const int mblk   = (Mtiles + 7) / 8;

    // ---- Layer 1 ----
    {
        dim3 grid((HEADS * HID) / 16, mblk);
        gat_gemm_wmma_f32<<<grid, 256, 0, stream>>>(x, W1, h1, Nn, IN, HEADS * HID);
    }
    {
        int t = Nn * HEADS;
        gat_attn_coef<<<(t + 255) / 256, 256, 0, stream>>>(h1, as1w, ad1w,
                                                           a_src1, a_dst1, Nn, HEADS, HID);
    }
    {
        int t = EP * HEADS;
        gat_edge_denom<<<(t + 255) / 256, 256, 0, stream>>>(ei, E, EP, a_src1, a_dst1,
                                                            den1, HEADS);
    }
    {
        int blocks = (EP + 7) / 8;         // 8 waves / block, wave per edge
        gat_edge_aggregate<<<blocks, 256, 0, stream>>>(ei, E, EP, a_src1, a_dst1,
                                                       den1, h1, agg1, HEADS, HID);
    }
    {
        int t = Nn * HEADS * HID;
        gat_bias_relu<<<(t + 255) / 256, 256, 0, stream>>>(agg1, b1, t, HEADS * HID);
    }

    // ---- Layer 2 ----
    {
        dim3 grid(OUT / 16, mblk);
        gat_gemm_wmma_f32<<<grid, 256, 0, stream>>>(agg1, W2, h2, Nn, HEADS * HID, OUT);
    }
    {
        int t = Nn;
        gat_attn_coef<<<(t + 255) / 256, 256, 0, stream>>>(h2, as2w, ad2w,
                                                           a_src2, a_dst2, Nn, 1, OUT);
    }
    {
        int t = EP;
        gat_edge_denom<<<(t + 255) / 256, 256, 0, stream>>>(ei, E, EP, a_src2, a_dst2,
                                                            den2, 1);
    }
    {
        int blocks = (EP + 7) / 8;
        gat_edge_aggregate<<<blocks, 256, 0, stream>>>(ei, E, EP, a_src2, a_dst2,
                                                       den2, h2, outF, 1, OUT);
    }
    {
        int t = Nn * OUT;
        gat_bias_add<<<(t + 255) / 256, 256, 0, stream>>>(outF, b2, t, OUT);
    }
}